// SAGEEncoder_3418793967880
// MI455X (gfx1250) — compile-verified
//
#include <hip/hip_runtime.h>
#include <hip/hip_bf16.h>

// GraphSAGE 2-layer encoder for MI455X (gfx1250, wave32).
// GEMMs: fp32 WMMA (v_wmma_f32_16x16x4_f32); weights double-buffered in LDS
// via GLOBAL_LOAD_ASYNC_TO_LDS_B128 with counted s_wait_asynccnt pipelining.

typedef float v2f __attribute__((ext_vector_type(2)));
typedef float v4f __attribute__((ext_vector_type(4)));
typedef float v8f __attribute__((ext_vector_type(8)));

// exact parameter types of __builtin_amdgcn_global_load_async_to_lds_b128
typedef int b128i __attribute__((vector_size(16)));
typedef __attribute__((address_space(1))) b128i* gb128_t;   // global (AS1)
typedef __attribute__((address_space(3))) b128i* lb128_t;   // LDS (AS3)

#define IN_C   128
#define HID_C  256
#define OUT_C  128
#define BN_EPS 1e-5f

#define KC      64      // K-chunk staged in LDS
#define BSTRIDE 72      // 64 cols + 8 pad: 2*72 % 64 == 16 -> half-waves hit
                        // disjoint LDS bank sets

#if __has_builtin(__builtin_amdgcn_global_load_async_to_lds_b128) && \
    __has_builtin(__builtin_amdgcn_s_wait_asynccnt)
#define USE_ASYNC_LDS 1
#else
#define USE_ASYNC_LDS 0
#endif

// ---------------------------------------------------------------- utilities

__global__ void zero_f32(float* __restrict__ p, long n) {
  long t = (long)blockIdx.x * blockDim.x + threadIdx.x;
  long stride = (long)gridDim.x * blockDim.x;
  for (; t < n; t += stride) p[t] = 0.0f;
}

__global__ void count_deg(const long long* __restrict__ dst,
                          float* __restrict__ cnt, int E) {
  int e = blockIdx.x * blockDim.x + threadIdx.x;
  if (e >= E) return;
  atomicAdd(&cnt[(int)dst[e]], 1.0f);
}

// scatter-add: one wave per edge, lanes stride channels (coalesced).
// All accumulators (25.6-51 MB) are L2-resident (192 MB) -> on-chip RMW.
template <int C>
__global__ __launch_bounds__(256)
void scatter_add(const long long* __restrict__ src,
                 const long long* __restrict__ dst,
                 const float* __restrict__ x,
                 float* __restrict__ agg, int E) {
  int wave = blockIdx.x * (blockDim.x >> 5) + (threadIdx.x >> 5);
  int lane = threadIdx.x & 31;
  if (wave >= E) return;
  long s = (long)src[wave];
  long d = (long)dst[wave];
  const float* xs = x + s * C;
  float* ad = agg + d * C;
#pragma unroll
  for (int c = lane; c < C; c += 32) {
    atomicAdd(&ad[c], xs[c]);
  }
}

__global__ void normalize_rows(float* __restrict__ agg,
                               const float* __restrict__ cnt,
                               long total, int logC) {
  long t = (long)blockIdx.x * blockDim.x + threadIdx.x;
  if (t >= total) return;
  float c = cnt[t >> logC];
  agg[t] *= 1.0f / fmaxf(c, 1.0f);
}

// ---------------------------------------------------------------- WMMA GEMM
// C[M x Ncols] = A1@W1 + A2@W2 + bias.
// Block = 8 waves = 128 rows sharing one 64-col weight strip, staged in LDS
// in KC-row chunks, double-buffered: chunk i+1's 4 async b128 loads per
// thread are issued before waiting (s_wait_asynccnt<=4 releases chunk i,
// in-order retirement). The flattened chunk sequence spans both A1@W1 and
// A2@W2 so the pipeline never drains mid-kernel.

__device__ __forceinline__ void stage_issue(const float* __restrict__ W,
                                            int k0, int n0, int Ncols,
                                            float* bs, int tid) {
  // 64 rows x 64 cols = 16KB; 256 threads x 4 rounds x 16B (issue only).
#pragma unroll
  for (int rnd = 0; rnd < 4; ++rnd) {
    int f  = tid + (rnd << 8);       // 0..1023
    int r  = f >> 4;                 // row 0..63
    int c4 = (f & 15) << 2;          // col 0,4,...,60
    const float* g = W + (long)(k0 + r) * Ncols + n0 + c4;
    float*       l = bs + r * BSTRIDE + c4;
#if USE_ASYNC_LDS
    __builtin_amdgcn_global_load_async_to_lds_b128((gb128_t)g, (lb128_t)l, 0, 0);
#else
    *(v4f*)l = *(const v4f*)g;       // global_load_b128 + ds_store_b128
#endif
  }
}

__device__ __forceinline__ v8f wmma4(v2f a, v2f b, v8f c) {
  return __builtin_amdgcn_wmma_f32_16x16x4_f32(false, a, false, b, (short)0, c,
                                               false, false);
}

__global__ __launch_bounds__(256)
void gemm_dual_wmma(const float* __restrict__ A1, const float* __restrict__ A2,
                    const float* __restrict__ W1, const float* __restrict__ W2,
                    const float* __restrict__ bias, float* __restrict__ C,
                    int M, int K1, int K2, int Ncols, int nShift) {
  __shared__ __align__(16) float bs[2][KC * BSTRIDE];  // 2 x 18 KB of 320 KB LDS
  const int tid   = threadIdx.x;
  const int lane  = tid & 31;
  const int half  = lane >> 4;
  const int l15   = lane & 15;
  const int wave  = tid >> 5;
  const int nStrip = blockIdx.x & ((1 << nShift) - 1);
  const int m0    = ((blockIdx.x >> nShift) << 7) + (wave << 4);  // 128 rows/block
  const int n0    = nStrip << 6;
  const bool active = (m0 < M);

  v8f z = {0.f, 0.f, 0.f, 0.f, 0.f, 0.f, 0.f, 0.f};
  v8f acc0 = z, acc1 = z, acc2 = z, acc3 = z;

  const int nch1 = K1 / KC;
  const int nch  = nch1 + K2 / KC;

  // prologue: start filling buffer 0 with chunk 0 (phase 1 always first)
  stage_issue(W1, 0, n0, Ncols, bs[0], tid);

  for (int i = 0; i < nch; ++i) {
    // issue next chunk into the other buffer, then wait only for chunk i.
    int nxt = i + 1;
    if (nxt < nch) {
      const float* Wn = (nxt < nch1) ? W1 : W2;
      int k0n = (nxt < nch1) ? nxt * KC : (nxt - nch1) * KC;
      stage_issue(Wn, k0n, n0, Ncols, bs[nxt & 1], tid);
#if USE_ASYNC_LDS
      __builtin_amdgcn_s_wait_asynccnt(4);   // chunk i retired, i+1 in flight
#endif
    } else {
#if USE_ASYNC_LDS
      __builtin_amdgcn_s_wait_asynccnt(0);
#endif
    }
    __syncthreads();                         // chunk i visible to all waves

    if (active) {                            // wave-uniform: EXEC all-1s (WMMA)
      const float* A = (i < nch1) ? A1 : A2;
      int K  = (i < nch1) ? K1 : K2;
      int k0 = (i < nch1) ? i * KC : (i - nch1) * KC;
      const float* Arow = A + (long)(m0 + l15) * K + k0 + half * 2;
      if (k0 + KC < K) __builtin_prefetch(Arow + KC, 0, 3);
      const float* B = bs[i & 1];
#pragma unroll 4
      for (int kk = 0; kk < KC; kk += 4) {
        v2f a;
        a.x = Arow[kk];
        a.y = Arow[kk + 1];
        const float* Bk = B + (kk + half * 2) * BSTRIDE + l15;
        v2f b0; b0.x = Bk[0];  b0.y = Bk[BSTRIDE + 0];
        v2f b1; b1.x = Bk[16]; b1.y = Bk[BSTRIDE + 16];
        v2f b2; b2.x = Bk[32]; b2.y = Bk[BSTRIDE + 32];
        v2f b3; b3.x = Bk[48]; b3.y = Bk[BSTRIDE + 48];
        acc0 = wmma4(a, b0, acc0);
        acc1 = wmma4(a, b1, acc1);
        acc2 = wmma4(a, b2, acc2);
        acc3 = wmma4(a, b3, acc3);
      }
    }
    __syncthreads();   // seal reads of bs[i&1] before it is re-staged at i+2
  }

  if (!active) return;
#pragma unroll
  for (int j = 0; j < 4; ++j) {
    v8f acc = (j == 0) ? acc0 : (j == 1) ? acc1 : (j == 2) ? acc2 : acc3;
    int col = n0 + j * 16 + l15;
    float bv = bias[col];
#pragma unroll
    for (int r = 0; r < 8; ++r) {
      int row = m0 + r + half * 8;
      C[(long)row * Ncols + col] = acc[r] + bv;
    }
  }
}

// ---------------------------------------------------------------- batchnorm

__global__ __launch_bounds__(256)
void bn_stats(const float* __restrict__ h, float* __restrict__ sum,
              float* __restrict__ sumsq, int N) {
  int c = threadIdx.x;                         // 0..255 == channel (coalesced)
  int rowsPerBlock = (N + gridDim.x - 1) / gridDim.x;
  int r0 = blockIdx.x * rowsPerBlock;
  int r1 = min(N, r0 + rowsPerBlock);
  float s = 0.f, ss = 0.f;
  for (int r = r0; r < r1; ++r) {
    float v = h[(long)r * HID_C + c];
    s += v;
    ss += v * v;
  }
  atomicAdd(&sum[c], s);
  atomicAdd(&sumsq[c], ss);
}

__global__ void bn_finalize(const float* __restrict__ sum,
                            const float* __restrict__ sumsq,
                            const float* __restrict__ gamma,
                            const float* __restrict__ beta,
                            float* __restrict__ scale,
                            float* __restrict__ shift, int N) {
  int c = threadIdx.x;
  float invN = 1.0f / (float)N;
  float mean = sum[c] * invN;
  float var  = sumsq[c] * invN - mean * mean;  // biased (BN train mode)
  float inv  = rsqrtf(var + BN_EPS);
  float sc   = gamma[c] * inv;
  scale[c] = sc;
  shift[c] = beta[c] - mean * sc;
}

__global__ void bn_relu(float* __restrict__ h,
                        const float* __restrict__ scale,
                        const float* __restrict__ shift, long total) {
  long t = (long)blockIdx.x * blockDim.x + threadIdx.x;
  if (t >= total) return;
  int c = (int)(t & (HID_C - 1));
  h[t] = fmaxf(h[t] * scale[c] + shift[c], 0.0f);
}

// ---------------------------------------------------------------- launcher

extern "C" void kernel_launch(void* const* d_in, const int* in_sizes, int n_in,
                              void* d_out, int out_size, void* d_ws, size_t ws_size,
                              hipStream_t stream) {
  const float*     x     = (const float*)d_in[0];
  const long long* eidx  = (const long long*)d_in[1];   // int64 [2, E]
  const float*     W1l   = (const float*)d_in[2];
  const float*     b1l   = (const float*)d_in[3];
  const float*     W1r   = (const float*)d_in[4];
  const float*     gamma = (const float*)d_in[5];
  const float*     beta  = (const float*)d_in[6];
  const float*     W2l   = (const float*)d_in[7];
  const float*     b2l   = (const float*)d_in[8];
  const float*     W2r   = (const float*)d_in[9];
  float*           out   = (float*)d_out;

  const int N = in_sizes[0] / IN_C;       // 50000
  const int E = in_sizes[1] / 2;          // 800000
  const long long* src = eidx;
  const long long* dst = eidx + E;

  // workspace layout (floats)
  float* ws    = (float*)d_ws;
  float* cnt   = ws;                              // N
  float* agg1  = cnt + N;                         // N*IN_C
  float* h     = agg1 + (long)N * IN_C;           // N*HID_C
  float* agg2  = h + (long)N * HID_C;             // N*HID_C
  float* bnsum = agg2 + (long)N * HID_C;          // HID_C
  float* bnsq  = bnsum + HID_C;                   // HID_C
  float* scale = bnsq + HID_C;                    // HID_C
  float* shift = scale + HID_C;                   // HID_C

  // --- zero accumulators (ws is not re-poisoned between replays) ---
  zero_f32<<<2048, 256, 0, stream>>>(cnt, (long)N + (long)N * IN_C);
  zero_f32<<<2048, 256, 0, stream>>>(agg2, (long)N * HID_C);
  zero_f32<<<2, 256, 0, stream>>>(bnsum, 2 * HID_C);

  // --- degrees (shared by both layers) ---
  count_deg<<<(E + 255) / 256, 256, 0, stream>>>(dst, cnt, E);

  // --- layer 1 aggregation ---
  scatter_add<IN_C><<<(E + 7) / 8, 256, 0, stream>>>(src, dst, x, agg1, E);
  {
    long tot = (long)N * IN_C;
    normalize_rows<<<(int)((tot + 255) / 256), 256, 0, stream>>>(agg1, cnt, tot, 7);
  }

  // --- layer 1 dual GEMM: h = agg1@W1l + x@W1r + b1l ---
  {
    int tilesM8 = (N / 16 + 7) / 8;    // 128-row blocks
    int nShift  = 2;                   // HID_C/64 = 4 strips
    gemm_dual_wmma<<<tilesM8 << nShift, 256, 0, stream>>>(
        agg1, x, W1l, W1r, b1l, h, N, IN_C, IN_C, HID_C, nShift);
  }

  // --- batchnorm (train) + ReLU ---
  bn_stats<<<512, 256, 0, stream>>>(h, bnsum, bnsq, N);
  bn_finalize<<<1, HID_C, 0, stream>>>(bnsum, bnsq, gamma, beta, scale, shift, N);
  {
    long tot = (long)N * HID_C;
    bn_relu<<<(int)((tot + 255) / 256), 256, 0, stream>>>(h, scale, shift, tot);
  }

  // --- layer 2 aggregation ---
  scatter_add<HID_C><<<(E + 7) / 8, 256, 0, stream>>>(src, dst, h, agg2, E);
  {
    long tot = (long)N * HID_C;
    normalize_rows<<<(int)((tot + 255) / 256), 256, 0, stream>>>(agg2, cnt, tot, 8);
  }

  // --- layer 2 dual GEMM: out = agg2@W2l + h@W2r + b2l ---
  {
    int tilesM8 = (N / 16 + 7) / 8;
    int nShift  = 1;                   // OUT_C/64 = 2 strips
    gemm_dual_wmma<<<tilesM8 << nShift, 256, 0, stream>>>(
        agg2, h, W2l, W2r, b2l, out, N, HID_C, HID_C, OUT_C, nShift);
  }
}